// OctreeConvTriton_91044716740959
// MI455X (gfx1250) — compile-verified
//
#include <hip/hip_runtime.h>
#include <hip/hip_bf16.h>

typedef __attribute__((ext_vector_type(2))) float v2f;
typedef __attribute__((ext_vector_type(8))) float v8f;

#define KTAPS 27
#define CIN   32
#define COUT  32
#define MTILE 32               // nodes per workgroup (2 x 16-row M-subtiles)

// ---------------------------------------------------------------------------
// Weight transpose: w[k][cin][cout] -> wt[k][cout][cin], so each lane's B
// fragment {W[k][kk][cout], W[k][kk+1][cout]} is one contiguous b64 load.
// ---------------------------------------------------------------------------
__global__ void octconv_transpose_w(const float* __restrict__ w,
                                    float* __restrict__ wt) {
    int i = blockIdx.x * 256 + threadIdx.x;
    if (i < KTAPS * CIN * COUT) {
        int k    = i / (CIN * COUT);
        int rem  = i - k * (CIN * COUT);
        int cout = rem / CIN;          // wt is [k][cout][cin]
        int cin  = rem - cout * CIN;
        wt[i] = w[(k * CIN + cin) * COUT + cout];
    }
}

// ---------------------------------------------------------------------------
// Main kernel: 32-node M-tile per workgroup (64 threads = 2 waves).
// Wave w computes cout[w*16 : w*16+16) for both 16-row M-subtiles, so each
// B fragment load feeds 2 WMMAs. Gathered rows for tap k+1 are staged into
// the spare LDS buffer with GLOBAL_LOAD_ASYNC_TO_LDS_B128 (ASYNCcnt) while
// the 16 V_WMMA_F32_16X16X4_F32 ops for tap k execute.
// ---------------------------------------------------------------------------
__global__ __launch_bounds__(64)
void octconv_wmma_f32(const float* __restrict__ data,
                      const float* __restrict__ wt,    // [k][cout][cin]
                      const float* __restrict__ bias,
                      const long long* __restrict__ neigh,
                      float* __restrict__ out,
                      int n_nodes) {
    __shared__ __align__(16) float  Alds[2][MTILE * CIN];  // 2 x 4KB gathered tile
    __shared__           long long  nbs[MTILE * KTAPS];    // tile neighbor rows

    const int t         = threadIdx.x;
    const int node_base = blockIdx.x * MTILE;

    // The tile's 32x27 neighbor block is contiguous: straight strided copy.
    {
        const long long* src = neigh + (long long)node_base * KTAPS;
        const int valid = (n_nodes - node_base) * KTAPS;   // tail guard
        for (int i = t; i < MTILE * KTAPS; i += 64)
            nbs[i] = (i < valid) ? src[i] : (long long)-1;
    }
    __syncthreads();

    // Async masked gather of tap k into LDS buffer `buf`.
    // 64 threads: thread t -> row r = t>>1 (0..31), half h = t&1 (64 bytes).
    // Valid rows: 4x async b128 direct to LDS (no VGPR round trip).
    // Empty rows: zero-filled via ds stores, so EXEC is all-1s at every WMMA.
    auto stage = [&](int k, int buf) {
        int r = t >> 1;
        int h = t & 1;
        long long row = nbs[r * KTAPS + k];
        float* lslot = &Alds[buf][r * CIN + h * 16];
        if (row >= 0) {
            unsigned laddr = (unsigned)(uintptr_t)lslot;
            unsigned long long gaddr =
                (unsigned long long)(uintptr_t)(data + (size_t)row * CIN + h * 16);
            asm volatile(
                "global_load_async_to_lds_b128 %0, %1, off\n\t"
                "global_load_async_to_lds_b128 %0, %1, off offset:16\n\t"
                "global_load_async_to_lds_b128 %0, %1, off offset:32\n\t"
                "global_load_async_to_lds_b128 %0, %1, off offset:48"
                :: "v"(laddr), "v"(gaddr) : "memory");
        } else {
            float4 z = make_float4(0.f, 0.f, 0.f, 0.f);
            float4* dst = (float4*)lslot;
            dst[0] = z; dst[1] = z; dst[2] = z; dst[3] = z;
        }
    };

    const int lane = t & 31;
    const int wv   = t >> 5;           // which 16-wide cout tile
    const int m    = lane & 15;        // row within subtile / D column
    const int kh   = lane >> 4;        // K-half select (ISA 16x4 f32 A layout)
    const int cout = wv * 16 + m;      // this lane's output channel

    v8f acc0 = {};                     // M-subtile 0 accumulator (8 VGPRs)
    v8f acc1 = {};                     // M-subtile 1 accumulator (8 VGPRs)

    stage(0, 0);
    asm volatile("s_wait_asynccnt 0x0" ::: "memory");
    __syncthreads();

    for (int k = 0; k < KTAPS; ++k) {
        const int buf = k & 1;
        if (k + 1 < KTAPS) stage(k + 1, buf ^ 1);   // overlaps WMMAs below

        const float* arow0 = &Alds[buf][m * CIN + 2 * kh];          // subtile 0
        const float* arow1 = &Alds[buf][(16 + m) * CIN + 2 * kh];   // subtile 1
        const float* brow  = wt + ((k * COUT + cout) * CIN + 2 * kh);
#pragma unroll
        for (int kb = 0; kb < CIN; kb += 4) {
            v2f b  = *(const v2f*)(brow + kb);    // one b64, feeds 2 WMMAs
            v2f a0 = *(const v2f*)(arow0 + kb);   // ds_load_b64
            v2f a1 = *(const v2f*)(arow1 + kb);
            acc0 = __builtin_amdgcn_wmma_f32_16x16x4_f32(
                false, a0, false, b, (short)0, acc0, false, false);
            acc1 = __builtin_amdgcn_wmma_f32_16x16x4_f32(
                false, a1, false, b, (short)0, acc1, false, false);
        }
        // Publish buf^1: drain our async gathers, then barrier.
        asm volatile("s_wait_asynccnt 0x0" ::: "memory");
        __syncthreads();
    }

    // D layout: VGPR j holds M=j (lanes 0-15) / M=j+8 (lanes 16-31), N=lane&15.
    const float bv = bias[cout];
#pragma unroll
    for (int j = 0; j < 8; ++j) {
        int n0 = node_base + j + 8 * kh;
        int n1 = node_base + 16 + j + 8 * kh;
        if (n0 < n_nodes) out[(long long)n0 * COUT + cout] = acc0[j] + bv;
        if (n1 < n_nodes) out[(long long)n1 * COUT + cout] = acc1[j] + bv;
    }
}

// ---------------------------------------------------------------------------
// Inputs (setup_inputs order): data[N,32] f32, weights[27,32,32] f32,
// bias[32] f32, neigh[N,27] int64. Output: [N,32] f32.
// ---------------------------------------------------------------------------
extern "C" void kernel_launch(void* const* d_in, const int* in_sizes, int n_in,
                              void* d_out, int out_size, void* d_ws, size_t ws_size,
                              hipStream_t stream) {
    const float*     data  = (const float*)d_in[0];
    const float*     w     = (const float*)d_in[1];
    const float*     bias  = (const float*)d_in[2];
    const long long* neigh = (const long long*)d_in[3];
    float*           out   = (float*)d_out;

    const int n_nodes = in_sizes[0] / CIN;
    float* wt = (float*)d_ws;   // 27*32*32*4 = 110,592 bytes of scratch

    octconv_transpose_w<<<(KTAPS * CIN * COUT + 255) / 256, 256, 0, stream>>>(w, wt);

    const int ntiles = (n_nodes + MTILE - 1) / MTILE;
    octconv_wmma_f32<<<ntiles, 64, 0, stream>>>(data, wt, bias, neigh, out, n_nodes);
}